// SpanEncoder_45174466019996
// MI455X (gfx1250) — compile-verified
//
#include <hip/hip_runtime.h>
#include <hip/hip_bf16.h>
#include <math.h>

#define T_TOK    2048
#define SDIM     1024
#define EDIM     1024
#define HID      150
#define HIDP     160      // HID padded to multiple of 16
#define MAXW     10
#define OUTCOLS  (2*SDIM + EDIM)   // 3072

typedef float v2f __attribute__((ext_vector_type(2)));
typedef float v4f __attribute__((ext_vector_type(4)));
typedef float v8f __attribute__((ext_vector_type(8)));

// ---------------------------------------------------------------------------
// Zero-padded weight copy: wp[Npad x Kpad] <- w[N x K]
// ---------------------------------------------------------------------------
__global__ void pad_weights(const float* __restrict__ w, float* __restrict__ wp,
                            int N, int K, int Npad, int Kpad) {
    int idx = blockIdx.x * blockDim.x + threadIdx.x;
    int total = Npad * Kpad;
    if (idx >= total) return;
    int n = idx / Kpad;
    int c = idx - n * Kpad;
    wp[idx] = (n < N && c < K) ? w[n * K + c] : 0.0f;
}

// ---------------------------------------------------------------------------
// C[M x Npad] = relu(A[M x K] * Wp[Npad x K]^T + bias)   (bias only for n<nBias)
// One wave (32 lanes) computes one 16x16 tile via V_WMMA_F32_16X16X4_F32.
//
// f32 WMMA layouts (wave32):
//   A 16x4 : lane&15 = M row, lane>>4 selects K pair {0,1} / {2,3} -> float2
//   B 4x16 : lane&15 = N col, lane>>4 selects K pair               -> float2
//   C 16x16: VGPR v, lanes0-15 -> (M=v, N=lane), lanes16-31 -> (M=v+8, N=lane-16)
// Both A (row-major M x K) and Wp (row-major N x K) load as contiguous float2.
// ---------------------------------------------------------------------------
__global__ __launch_bounds__(32)
void gemm_relu_wmma(const float* __restrict__ A,
                    const float* __restrict__ Wp,
                    const float* __restrict__ bias, int nBias,
                    float* __restrict__ C,
                    int K, int Npad) {
    const int lane = threadIdx.x;          // 0..31
    const int r    = lane & 15;            // row of A / col of W within tile
    const int half = lane >> 4;            // K-pair select
    const int koff = half * 2;

    const int m0 = blockIdx.x * 16;
    const int n0 = blockIdx.y * 16;

    const float* aRow = A  + (size_t)(m0 + r) * K;
    const float* bRow = Wp + (size_t)(n0 + r) * K;

    v8f acc = {};
    for (int k = 0; k < K; k += 4) {
        v2f av = *(const v2f*)(aRow + k + koff);
        v2f bv = *(const v2f*)(bRow + k + koff);
        acc = __builtin_amdgcn_wmma_f32_16x16x4_f32(
            /*neg_a=*/false, av, /*neg_b=*/false, bv,
            /*c_mod=*/(short)0, acc, /*reuse_a=*/false, /*reuse_b=*/false);
    }

    const int ncol = n0 + r;
    const float b = (ncol < nBias) ? bias[ncol] : 0.0f;
#pragma unroll
    for (int v = 0; v < 8; ++v) {
        int m = m0 + v + 8 * half;
        float val = acc[v] + b;
        val = fmaxf(val, 0.0f);            // relu (both layers use it)
        C[(size_t)m * Npad + ncol] = val;
    }
}

// ---------------------------------------------------------------------------
// attns[t] = dot(h2[t, 0:150], w3) + b3     (one wave per token)
// ---------------------------------------------------------------------------
__global__ __launch_bounds__(32)
void attn_score(const float* __restrict__ h2, const float* __restrict__ w3,
                const float* __restrict__ b3, float* __restrict__ attns) {
    const int t    = blockIdx.x;
    const int lane = threadIdx.x;
    float s = 0.0f;
    for (int i = lane; i < HID; i += 32)
        s += h2[(size_t)t * HIDP + i] * w3[i];
#pragma unroll
    for (int off = 16; off > 0; off >>= 1)
        s += __shfl_down(s, off, 32);
    if (lane == 0) attns[t] = s + b3[0];
}

// ---------------------------------------------------------------------------
// Span assembly: one 256-thread block per output row.
// Row = [states[w], states[w+n-1], softmax(attns[w..w+n-1]) . embeds[w..w+n-1]]
// Each thread owns one float4 column group (256 * 4 = 1024 cols per section).
//
// Output is 251 MB written exactly once -> store with TH=NT (non-temporal)
// so the write stream does not evict the L2-resident states/embeds (16 MB),
// which absorb ~600 MB of logical re-reads. Input loads stay temporal.
// ---------------------------------------------------------------------------
__global__ __launch_bounds__(256)
void span_kernel(const float* __restrict__ embeds,
                 const float* __restrict__ states,
                 const float* __restrict__ attns,
                 float* __restrict__ out) {
    const int row = blockIdx.x;

    // Decode span width n and start w from flat row index.
    int n = 1, base = 0;
    for (int m = 1; m <= MAXW; ++m) {
        int cnt = T_TOK - m + 1;
        if (row < base + cnt) { n = m; break; }
        base += cnt;
    }
    const int w = row - base;

    // Per-window softmax over <= 10 scalars (attns is tiny & cache-hot).
    float a[MAXW];
    float mx = -INFINITY;
    for (int j = 0; j < n; ++j) { a[j] = attns[w + j]; mx = fmaxf(mx, a[j]); }
    float s = 0.0f;
    for (int j = 0; j < n; ++j) { a[j] = __expf(a[j] - mx); s += a[j]; }
    const float inv = 1.0f / s;

    const int t = threadIdx.x;                         // 0..255 float4 groups
    const v4f* st4 = (const v4f*)states;
    const v4f* em4 = (const v4f*)embeds;
    v4f* o = (v4f*)(out + (size_t)row * OUTCOLS);

    // start / end states: copy through (temporal loads, NT stores)
    v4f s0 = st4[(size_t)w * 256 + t];
    v4f s1 = st4[(size_t)(w + n - 1) * 256 + t];
    __builtin_nontemporal_store(s0, o + t);
    __builtin_nontemporal_store(s1, o + 256 + t);

    v4f acc = {0.f, 0.f, 0.f, 0.f};
    for (int j = 0; j < n; ++j) {
        const float aj = a[j] * inv;
        const v4f e = em4[(size_t)(w + j) * 256 + t];
        acc += aj * e;
    }
    __builtin_nontemporal_store(acc, o + 512 + t);     // pooled embed
}

// ---------------------------------------------------------------------------
extern "C" void kernel_launch(void* const* d_in, const int* in_sizes, int n_in,
                              void* d_out, int out_size, void* d_ws, size_t ws_size,
                              hipStream_t stream) {
    const float* embeds = (const float*)d_in[0];   // T x EDIM
    const float* states = (const float*)d_in[1];   // T x SDIM
    const float* w1     = (const float*)d_in[2];   // HID x SDIM
    const float* b1     = (const float*)d_in[3];   // HID
    const float* w2     = (const float*)d_in[4];   // HID x HID
    const float* b2     = (const float*)d_in[5];   // HID
    const float* w3     = (const float*)d_in[6];   // 1 x HID
    const float* b3     = (const float*)d_in[7];   // 1
    float* out = (float*)d_out;

    // Workspace layout (floats)
    float* ws    = (float*)d_ws;
    float* w1p   = ws;                               // HIDP * SDIM
    float* w2p   = w1p + (size_t)HIDP * SDIM;        // HIDP * HIDP
    float* h1    = w2p + (size_t)HIDP * HIDP;        // T * HIDP
    float* h2    = h1  + (size_t)T_TOK * HIDP;       // T * HIDP
    float* attns = h2  + (size_t)T_TOK * HIDP;       // T

    // 1) Zero-pad the MLP weights so padded K/N lanes contribute exactly 0.
    {
        int tot1 = HIDP * SDIM;
        pad_weights<<<(tot1 + 255) / 256, 256, 0, stream>>>(w1, w1p, HID, SDIM, HIDP, SDIM);
        int tot2 = HIDP * HIDP;
        pad_weights<<<(tot2 + 255) / 256, 256, 0, stream>>>(w2, w2p, HID, HID, HIDP, HIDP);
    }

    // 2) h1 = relu(states @ w1^T + b1)   -- WMMA f32 16x16x4
    {
        dim3 grid(T_TOK / 16, HIDP / 16);
        gemm_relu_wmma<<<grid, 32, 0, stream>>>(states, w1p, b1, HID, h1, SDIM, HIDP);
    }

    // 3) h2 = relu(h1 @ w2^T + b2)       -- WMMA f32 16x16x4 (padded K=160)
    {
        dim3 grid(T_TOK / 16, HIDP / 16);
        gemm_relu_wmma<<<grid, 32, 0, stream>>>(h1, w2p, b2, HID, h2, HIDP, HIDP);
    }

    // 4) attns = h2 @ w3^T + b3
    attn_score<<<T_TOK, 32, 0, stream>>>(h2, w3, b3, attns);

    // 5) Span assembly: 20435 rows x 3072 cols (bandwidth-bound NT stores).
    int n_rows = 0;
    for (int m = 1; m <= MAXW; ++m) n_rows += T_TOK - m + 1;   // 20435
    span_kernel<<<n_rows, 256, 0, stream>>>(embeds, states, attns, out);

    (void)in_sizes; (void)n_in; (void)out_size; (void)ws_size;
}